// ClusterPooling_28913719837075
// MI455X (gfx1250) — compile-verified
//
#include <hip/hip_runtime.h>
#include <hip/hip_bf16.h>
#include <stdint.h>

// ---------------------------------------------------------------------------
// ClusterPooling for MI455X (gfx1250).
// Memory-bound op (~0.77 GB moved -> ~33us at 23.3 TB/s). No matmul FLOPs, so
// no WMMA; the CDNA5-specific path is the Tensor Data Mover: each block DMAs a
// 32x256-float tile of x into LDS (TENSORcnt-tracked), then scatter-adds with
// hardware global_atomic_add_f32 into a 102MB accumulator that lives in the
// 192MB L2.
// ---------------------------------------------------------------------------

#define C_CH 256
#define ROWS 32

typedef unsigned int       u32;
typedef unsigned long long u64;
typedef __attribute__((ext_vector_type(4))) u32 v4u;
typedef __attribute__((ext_vector_type(8))) int v8i;
typedef __attribute__((ext_vector_type(4))) int v4i;

#if defined(__gfx1250__) && __has_builtin(__builtin_amdgcn_tensor_load_to_lds) && \
    __has_builtin(__builtin_amdgcn_s_wait_tensorcnt)
#define HAVE_TDM 1
#else
#define HAVE_TDM 0
#endif

__device__ __forceinline__ void atomAddF(float* p, float v) {
#if defined(__HIP_DEVICE_COMPILE__)
  unsafeAtomicAdd(p, v);   // hardware global_atomic_add_f32 (no CAS loop)
#else
  (void)p; (void)v;
#endif
}

// ---- 1. zero the accumulators (must run every call; atomics accumulate) ----
__global__ void cp_init_kernel(float* __restrict__ xout, float* __restrict__ counts,
                               u64 nx, int M) {
  u64 i = (u64)blockIdx.x * blockDim.x + threadIdx.x;
  u64 stride = (u64)gridDim.x * blockDim.x;
  for (u64 j = i; j < nx; j += stride) xout[j] = 0.0f;
  for (u64 j = i; j < (u64)M; j += stride) counts[j] = 0.0f;
}

// ---- 2. per-cluster counts ----
__global__ void cp_count_kernel(const int* __restrict__ cluster,
                                float* __restrict__ counts, int N) {
  int i = blockIdx.x * blockDim.x + threadIdx.x;
  if (i < N) atomAddF(&counts[cluster[i]], 1.0f);
}

// ---- 3. scatter-add: TDM tile load -> LDS -> f32 atomics into L2 ----
__global__ __launch_bounds__(256)
void cp_scatter_kernel(const float* __restrict__ x,
                       const int* __restrict__ cluster,
                       float* __restrict__ xout, int N) {
  __shared__ float tile[ROWS * C_CH];          // 32 KB
  const int rowStart = blockIdx.x * ROWS;
  int rows = N - rowStart;
  if (rows <= 0) return;
  if (rows > ROWS) rows = ROWS;

#if HAVE_TDM
  if (threadIdx.x < 32) {
    // Build Tensor DMA Descriptor (D#): 2D tile [rows x 256] of 4-byte elems.
    const u64 gaddr = (u64)(uintptr_t)(x + (u64)rowStart * C_CH);
    const u32 laddr = (u32)(uintptr_t)(&tile[0]);      // low 32 bits = LDS byte addr

    v4u g0;
    g0.x = 1u;                                          // count=1 (valid), user mode
    g0.y = laddr;                                       // lds_addr [63:32]
    g0.z = (u32)gaddr;                                  // global_addr [95:64]
    g0.w = (u32)((gaddr >> 32) & 0x1FFFFFFull)          // global_addr [120:96]
         | (2u << 30);                                  // type=2 ("image")

    v8i g1;
    g1[0] = (int)(2u << 16);                            // data_size=2 (4B), mask=0
    g1[1] = (int)((u32)C_CH << 16);                     // tensor_dim0[15:0]=256
    g1[2] = (int)(((u32)rows & 0xFFFFu) << 16);         // dim0 hi=0 | tensor_dim1 lo=rows
    g1[3] = (int)((u32)C_CH << 16);                     // dim1 hi=0 | tile_dim0=256
    g1[4] = (int)((u32)rows & 0xFFFFu);                 // tile_dim1=rows | tile_dim2=0
    g1[5] = (int)C_CH;                                  // tensor_dim0_stride=256
    g1[6] = 0;
    g1[7] = 0;

    v4i gz4 = {0, 0, 0, 0};                             // groups 2/3 unused (2D)
    v8i gz8 = {0, 0, 0, 0, 0, 0, 0, 0};
    __builtin_amdgcn_tensor_load_to_lds(g0, g1, gz4, gz4, gz8, 0);
    __builtin_amdgcn_s_wait_tensorcnt(0);
  }
  __syncthreads();
#else
  {
    const float4* src = (const float4*)(x + (u64)rowStart * C_CH);
    float4* dst4 = (float4*)tile;
    const int n4 = rows * (C_CH / 4);
    for (int i = threadIdx.x; i < n4; i += blockDim.x) dst4[i] = src[i];
  }
  __syncthreads();
#endif

  // 64 threads cover one row (float4 each); 256 threads = 4 rows in flight.
  const int lane = threadIdx.x & 63;
  const int rsub = threadIdx.x >> 6;
  for (int r = rsub; r < rows; r += 4) {
    const int cl = cluster[rowStart + r];
    const float4 v = *(const float4*)&tile[r * C_CH + lane * 4];
    float* dst = xout + (u64)cl * C_CH + (u64)(lane * 4);
    atomAddF(dst + 0, v.x);
    atomAddF(dst + 1, v.y);
    atomAddF(dst + 2, v.z);
    atomAddF(dst + 3, v.w);
  }
}

// ---- 4. divide sums by counts (mean) ----
__global__ void cp_finalize_kernel(float* __restrict__ xout,
                                   const float* __restrict__ counts, int M) {
  float4* x4 = (float4*)xout;
  const u64 n4 = (u64)M * (C_CH / 4);
  u64 i = (u64)blockIdx.x * blockDim.x + threadIdx.x;
  const u64 stride = (u64)gridDim.x * blockDim.x;
  for (u64 j = i; j < n4; j += stride) {
    const int row = (int)(j >> 6);                       // 64 float4 per row
    const float inv = 1.0f / fmaxf(counts[row], 1.0f);
    float4 v = x4[j];
    v.x *= inv; v.y *= inv; v.z *= inv; v.w *= inv;
    x4[j] = v;
  }
}

// ---- 5a. pos/batch gathers ----
__global__ void cp_gather_kernel(const float* __restrict__ pos,
                                 const int* __restrict__ batch,
                                 const int* __restrict__ sample,
                                 float* __restrict__ pos_out,
                                 float* __restrict__ batch_out, int M) {
  int m = blockIdx.x * blockDim.x + threadIdx.x;
  if (m < M) {
    const int s = sample[m];
    pos_out[m * 3 + 0] = pos[s * 3 + 0];
    pos_out[m * 3 + 1] = pos[s * 3 + 1];
    pos_out[m * 3 + 2] = pos[s * 3 + 2];
    batch_out[m] = (float)batch[s];
  }
}

// ---- 5b. edge_index pass-through (int -> float), streaming/NT ----
__global__ void cp_edge_kernel(const int* __restrict__ e, float* __restrict__ out, int n) {
  u64 i = (u64)blockIdx.x * blockDim.x + threadIdx.x;
  const u64 stride = (u64)gridDim.x * blockDim.x;
  for (u64 j = i; j < (u64)n; j += stride) {
    const int v = __builtin_nontemporal_load(&e[j]);
    __builtin_nontemporal_store((float)v, &out[j]);
  }
}

extern "C" void kernel_launch(void* const* d_in, const int* in_sizes, int n_in,
                              void* d_out, int out_size, void* d_ws, size_t ws_size,
                              hipStream_t stream) {
  const float* x       = (const float*)d_in[0];   // [N, 256]
  const float* pos     = (const float*)d_in[1];   // [N, 3]
  const int*   batch   = (const int*)d_in[2];     // [N]
  const int*   cluster = (const int*)d_in[3];     // [N]
  const int*   sample  = (const int*)d_in[4];     // [M]
  const int*   edge    = (const int*)d_in[5];     // [2, E]

  const int N  = in_sizes[2];
  const int M  = in_sizes[4];
  const int E2 = in_sizes[5];                     // 2*E

  float* xout      = (float*)d_out;               // [M, 256]
  float* pos_out   = xout + (u64)M * C_CH;        // [M, 3]
  float* batch_out = pos_out + (u64)M * 3;        // [M]
  float* edge_out  = batch_out + M;               // [2*E]
  float* counts    = (float*)d_ws;                // [M] scratch

  const u64 nx = (u64)M * C_CH;

  cp_init_kernel<<<4096, 256, 0, stream>>>(xout, counts, nx, M);
  cp_count_kernel<<<(N + 255) / 256, 256, 0, stream>>>(cluster, counts, N);
  cp_scatter_kernel<<<(N + ROWS - 1) / ROWS, 256, 0, stream>>>(x, cluster, xout, N);
  cp_finalize_kernel<<<4096, 256, 0, stream>>>(xout, counts, M);
  cp_gather_kernel<<<(M + 255) / 256, 256, 0, stream>>>(pos, batch, sample,
                                                        pos_out, batch_out, M);
  cp_edge_kernel<<<(E2 + 255) / 256, 256, 0, stream>>>(edge, edge_out, E2);
}